// Encoder_19232863552191
// MI455X (gfx1250) — compile-verified
//
#include <hip/hip_runtime.h>
#include <cstdint>

// Elman RNN final-hidden scan for MI455X (gfx1250).
// Memory-bound: streams 128 MiB of x once. Uses CDNA5 async global->LDS DMA
// (ASYNCcnt path) to turn 1KiB-strided per-lane rows into coalesced loads,
// double-buffered over 16-timestep chunks; recurrence uses native v_tanh_f32.

#define BLOCK   128
#define TSTEPS  128
#define CT      16                  // timesteps per chunk
#define NCHUNK  (TSTEPS / CT)       // 8
#define ROWB    (CT * 8)            // 128 bytes of x per row per chunk
#define STRIDE  (ROWB + 8)          // 136 B = 34 dwords, 17 x 8B (odd) -> conflict-free reads
#define BUFB    (BLOCK * STRIDE)    // 17408 bytes per LDS buffer

static __device__ __forceinline__ float fast_tanh(float v) {
#if __has_builtin(__builtin_amdgcn_tanhf)
  return __builtin_amdgcn_tanhf(v);
#else
  float r;
  asm volatile("v_tanh_f32 %0, %1" : "=v"(r) : "v"(v));
  return r;
#endif
}

__global__ __launch_bounds__(BLOCK) void elman_scan_kernel(
    const float* __restrict__ x, const float* __restrict__ Wih,
    const float* __restrict__ bih, const float* __restrict__ Whh,
    const float* __restrict__ bhh, float* __restrict__ out) {
  __shared__ __align__(16) char smem[2 * BUFB];

  const int tid = threadIdx.x;
  // This block's 128 rows of x; each row is T*2 floats = 1024 bytes.
  const float* xblk = x + (size_t)blockIdx.x * (size_t)BLOCK * TSTEPS * 2;

  const unsigned ldsBase = (unsigned)(uintptr_t)(void*)&smem[0];
  const int rsub = tid >> 4;             // 0..7
  const unsigned joff = (unsigned)(tid & 15) * 8u;

  // Uniform tiny weights (4+4+2+2 floats); L2-resident after first block.
  const float w00 = Wih[0], w01 = Wih[1], w10 = Wih[2], w11 = Wih[3];
  const float u00 = Whh[0], u01 = Whh[1], u10 = Whh[2], u11 = Whh[3];
  const float c0 = bih[0] + bhh[0], c1 = bih[1] + bhh[1];

  // Issue one chunk's coalesced async loads: 16 x b64 per lane.
  // Lane pattern per instruction: 2 rows x 16 consecutive 8B = 2 full
  // 128B cachelines of global memory.
  auto issue_chunk = [&](int c, int p) {
    const unsigned bufBase = ldsBase + (unsigned)p * BUFB;
    const unsigned gchunk = (unsigned)c * ROWB;
#pragma unroll
    for (int i = 0; i < 16; ++i) {
      const int row = i * 8 + rsub;
      const unsigned lds = bufBase + (unsigned)row * STRIDE + joff;
      const unsigned goff = (unsigned)row * (TSTEPS * 8u) + gchunk + joff;
      asm volatile("global_load_async_to_lds_b64 %0, %1, %2"
                   :: "v"(lds), "v"(goff), "s"(xblk)
                   : "memory");
    }
  };

  float h0 = 0.0f, h1 = 0.0f;

  issue_chunk(0, 0);

  for (int c = 0; c < NCHUNK; ++c) {
    const int p = c & 1;
    if (c + 1 < NCHUNK) {
      issue_chunk(c + 1, p ^ 1);
      // chunk c done (in-order completion) while chunk c+1 stays in flight
      asm volatile("s_wait_asynccnt 16" ::: "memory");
    } else {
      asm volatile("s_wait_asynccnt 0" ::: "memory");
    }
    __syncthreads();  // publish all waves' DMA writes before cross-wave reads

    const char* rowp = smem + p * BUFB + tid * STRIDE;
#pragma unroll
    for (int j = 0; j < CT; ++j) {
      const float2 xv = *(const float2*)(rowp + j * 8);
      const float p0 =
          fmaf(u00, h0, fmaf(u01, h1, fmaf(w00, xv.x, fmaf(w01, xv.y, c0))));
      const float p1 =
          fmaf(u10, h0, fmaf(u11, h1, fmaf(w10, xv.x, fmaf(w11, xv.y, c1))));
      h0 = fast_tanh(p0);
      h1 = fast_tanh(p1);
    }
    __syncthreads();  // release this buffer before it is overwritten (c+2)
  }

  const unsigned b = blockIdx.x * BLOCK + tid;
  ((float2*)out)[b] = make_float2(h0, h1);  // coalesced 8B stores
}

extern "C" void kernel_launch(void* const* d_in, const int* in_sizes, int n_in,
                              void* d_out, int out_size, void* d_ws, size_t ws_size,
                              hipStream_t stream) {
  const float* x   = (const float*)d_in[0];
  const float* Wih = (const float*)d_in[1];
  const float* bih = (const float*)d_in[2];
  const float* Whh = (const float*)d_in[3];
  const float* bhh = (const float*)d_in[4];
  float* out = (float*)d_out;

  const int Bsz = in_sizes[0] / (TSTEPS * 2);  // 131072
  dim3 grid(Bsz / BLOCK);                      // 1024 blocks of 128 (4 waves)
  elman_scan_kernel<<<grid, BLOCK, 0, stream>>>(x, Wih, bih, Whh, bhh, out);
}